// SSDHead_23287312679611
// MI455X (gfx1250) — compile-verified
//
#include <hip/hip_runtime.h>
#include <hip/hip_bf16.h>
#include <math.h>

typedef __attribute__((ext_vector_type(2))) float        v2f;
typedef __attribute__((ext_vector_type(8))) float        v8f;
typedef __attribute__((ext_vector_type(4))) unsigned int v4u;
typedef __attribute__((ext_vector_type(8))) int          v8i;
typedef __attribute__((ext_vector_type(4))) int          v4i;

#define NCLS        91
#define TILE_ROWS   64
#define ROW_STRIDE  92          // 91 + 1 pad (TDM OOB zero-fill supplies col 91)
#define FUSED_BLOCK 128         // 4 waves * 16 rows = 64 rows per tile
#define NMS_BLOCKS  512
#define NMS_THREADS 256
#define DETS        100

#define SCORE_THRESH 0.05f
#define NMS_THRESH   0.5f
#define MIN_SIZE     0.01f
#define BBOX_CLIP    4.135166556742356f   // log(1000/16)

// ---- workspace layout (bytes) ----
#define WS_CNT    0                         // 1 u32: compacted candidate count
#define WS_BEST   64                        // f[0..3]=box, i[5]=label, i[6]=active
#define WS_PS     128                       // float[NMS_BLOCKS] partial best score
#define WS_PORIG  (WS_PS    + 4*NMS_BLOCKS) // int[NMS_BLOCKS] partial best orig row
#define WS_PPOS   (WS_PORIG + 4*NMS_BLOCKS) // int[NMS_BLOCKS] partial best cand pos
#define WS_CSCORE 8192                      // float[N]
#define WS_CLAB   (WS_CSCORE + 4*200000)    // int[N]
#define WS_CORIG  (WS_CLAB   + 4*200000)    // int[N]
#define WS_CBOX   (WS_CORIG  + 4*200000)    // float4[N] (16B aligned)

// ---------------------------------------------------------------------------
// TDM: issue a 2-D tensor_load_to_lds of a [rows x 91] f32 tile into LDS with
// tile_dim0=92 (col 91 zero-filled via OOB) and rows clamped by tensor_dim1.
// Descriptor packing per CDNA5 ISA §8.3 / §8.4. 6-arg builtin form
// (groups 0..3 + zero-filled trailing group + cache policy).
// ---------------------------------------------------------------------------
__device__ __forceinline__ void tdm_load_tile(const float* gsrc,
                                              unsigned lds_off,
                                              unsigned rows_rem) {
  unsigned long long ga = (unsigned long long)(size_t)gsrc;
  v4u g0;
  g0.x = 1u;                                            // count=1 (valid user D#)
  g0.y = lds_off;                                       // lds_addr (bytes)
  g0.z = (unsigned)(ga & 0xffffffffu);                  // global_addr[31:0]
  g0.w = (unsigned)((ga >> 32) & 0x01ffffffu) | (2u << 30);  // addr[56:32], type=2
  v8i g1;
  g1[0] = (int)(2u << 16);                              // data_size=4B; mask=0
  g1[1] = (int)((unsigned)NCLS << 16);                  // tensor_dim0[15:0]=91
  g1[2] = (int)((rows_rem & 0xffffu) << 16);            // dim0 hi=0; tensor_dim1 lo
  g1[3] = (int)((rows_rem >> 16) | ((unsigned)ROW_STRIDE << 16)); // dim1 hi; tile_dim0=92
  g1[4] = (int)TILE_ROWS;                               // tile_dim1=64; tile_dim2=0
  g1[5] = (int)NCLS;                                    // tensor_dim0_stride=91
  g1[6] = 0;
  g1[7] = 0;
  v4i g2 = {0, 0, 0, 0};
  v4i g3 = {0, 0, 0, 0};
  v8i g4 = {0, 0, 0, 0, 0, 0, 0, 0};
  __builtin_amdgcn_tensor_load_to_lds(g0, g1, g2, g3, g4, 0);
}

// ---------------------------------------------------------------------------
// Fused: TDM-staged softmax(top1/argmax) + box decode/clip + compaction.
// One wave handles 16 rows; lane L owns row (L&15), K half (L>>4).
// Row Σexp computed on the matrix pipe: 23x V_WMMA_F32_16X16X4_F32 vs ones.
// ---------------------------------------------------------------------------
__global__ __launch_bounds__(FUSED_BLOCK) void fused_decode_softmax(
    const float* __restrict__ logits, const float* __restrict__ breg,
    const float* __restrict__ props, const int* __restrict__ img_h,
    const int* __restrict__ img_w, unsigned char* __restrict__ ws,
    int N, int numTiles) {
  __shared__ float tiles[2][TILE_ROWS * ROW_STRIDE];   // 47,104 B

  unsigned* cnt   = (unsigned*)(ws + WS_CNT);
  float*    csc   = (float*)(ws + WS_CSCORE);
  int*      clab  = (int*)(ws + WS_CLAB);
  int*      corig = (int*)(ws + WS_CORIG);
  float4*   cbox  = (float4*)(ws + WS_CBOX);

  const float imH = (float)img_h[0];
  const float imW = (float)img_w[0];

  const int tid  = threadIdx.x;
  const int wave = tid >> 5;
  const int L    = tid & 31;
  const int r    = L & 15;
  const int half = L >> 4;
  const int j    = L & 7;
  const bool useOwn = (L < 8) || (L >= 24);
  const v2f ones = {1.0f, 1.0f};

  int t = (int)blockIdx.x;
  if (t < numTiles && tid == 0) {
    tdm_load_tile(logits + (size_t)t * TILE_ROWS * NCLS,
                  (unsigned)(size_t)(void*)&tiles[0][0],
                  (unsigned)(N - t * TILE_ROWS));
  }
  int buf = 0;
  while (t < numTiles) {
    const int nt = t + (int)gridDim.x;
    if (nt < numTiles) {
      if (tid == 0) {
        tdm_load_tile(logits + (size_t)nt * TILE_ROWS * NCLS,
                      (unsigned)(size_t)(void*)&tiles[buf ^ 1][0],
                      (unsigned)(N - nt * TILE_ROWS));
        __builtin_amdgcn_s_wait_tensorcnt(1);   // oldest (current tile) done
      }
    } else {
      if (tid == 0) __builtin_amdgcn_s_wait_tensorcnt(0);
    }
    __syncthreads();

    const float* rowp = &tiles[buf][(wave * 16 + r) * ROW_STRIDE];

    // pass 1: per-lane max/argmax over k = 4c + 2*half + {0,1}
    float m = -3.0e38f;
    int   am = 0;
#pragma unroll
    for (int c = 0; c < 23; ++c) {
      const int k0 = 4 * c + half * 2;
      const float v0 = rowp[k0];
      const float v1 = rowp[k0 + 1];
      if (v0 > m) { m = v0; am = k0; }
      if ((k0 + 1) < NCLS && v1 > m) { m = v1; am = k0 + 1; }
    }
    {  // combine with partner half of the row (first-index tie-break)
      const float pm = __shfl_xor(m, 16);
      const int   pa = __shfl_xor(am, 16);
      if (pm > m || (pm == m && pa < am)) { m = pm; am = pa; }
    }

    // pass 2: Σexp(l - m) on the XDL pipe (f32 WMMA, exact f32 accumulate)
    v8f acc = {0.f, 0.f, 0.f, 0.f, 0.f, 0.f, 0.f, 0.f};
#pragma unroll
    for (int c = 0; c < 23; ++c) {
      const int k0 = 4 * c + half * 2;
      const float e0 = __expf(rowp[k0] - m);
      const float e1 = ((k0 + 1) < NCLS) ? __expf(rowp[k0 + 1] - m) : 0.0f;
      v2f a = {e0, e1};
      acc = __builtin_amdgcn_wmma_f32_16x16x4_f32(false, a, false, ones,
                                                  (short)0, acc, false, false);
    }
    // extract row (L&15) sum from the 16x16 C tile
    const float a01 = (j & 1) ? acc[1] : acc[0];
    const float a23 = (j & 1) ? acc[3] : acc[2];
    const float a45 = (j & 1) ? acc[5] : acc[4];
    const float a67 = (j & 1) ? acc[7] : acc[6];
    const float a03 = (j & 2) ? a23 : a01;
    const float a47 = (j & 2) ? a67 : a45;
    const float own = (j & 4) ? a47 : a03;
    const float part = __shfl_xor(own, 16);
    const float ssum = useOwn ? own : part;
    const float top1 = 1.0f / ssum;          // exp(m-m)/Σ

    if (L < 16) {
      const int row = t * TILE_ROWS + wave * 16 + r;
      if (row < N) {
        const float4 rc = ((const float4*)breg)[row];
        const float4 pb = ((const float4*)props)[row];
        const float w  = pb.z - pb.x, h = pb.w - pb.y;
        const float cx = pb.x + 0.5f * w, cy = pb.y + 0.5f * h;
        const float dx = rc.x / 10.0f, dy = rc.y / 10.0f;
        const float dw = fminf(rc.z / 5.0f, BBOX_CLIP);
        const float dh = fminf(rc.w / 5.0f, BBOX_CLIP);
        const float pcx = dx * w + cx, pcy = dy * h + cy;
        const float pw = __expf(dw) * w, ph = __expf(dh) * h;
        float x1 = pcx - 0.5f * pw, y1 = pcy - 0.5f * ph;
        float x2 = pcx + 0.5f * pw, y2 = pcy + 0.5f * ph;
        x1 = fminf(fmaxf(x1, 0.0f), imW);
        x2 = fminf(fmaxf(x2, 0.0f), imW);
        y1 = fminf(fmaxf(y1, 0.0f), imH);
        y2 = fminf(fmaxf(y2, 0.0f), imH);
        const bool valid = (top1 > SCORE_THRESH) &&
                           ((x2 - x1) >= MIN_SIZE) && ((y2 - y1) >= MIN_SIZE);
        if (valid) {
          const unsigned p = atomicAdd(cnt, 1u);
          csc[p]   = top1;
          clab[p]  = am;
          corig[p] = row;
          cbox[p]  = make_float4(x1, y1, x2, y2);
        }
      }
    }
    __syncthreads();         // done reading tiles[buf] before it is re-issued
    t = nt;
    buf ^= 1;
  }
}

// ---------------------------------------------------------------------------
// NMS round, part 1: suppress vs current best (same label, IoU > 0.5), then
// per-block partial argmax with (score desc, orig-row asc) tie-break.
// ---------------------------------------------------------------------------
__global__ __launch_bounds__(NMS_THREADS) void nms_sweep(unsigned char* ws) {
  const unsigned K   = *(const unsigned*)(ws + WS_CNT);
  float*        csc  = (float*)(ws + WS_CSCORE);
  const int*    clab = (const int*)(ws + WS_CLAB);
  const int*    cor  = (const int*)(ws + WS_CORIG);
  const float4* cbx  = (const float4*)(ws + WS_CBOX);
  const float*  bf   = (const float*)(ws + WS_BEST);
  const int*    bi   = (const int*)(ws + WS_BEST);

  const float bx1 = bf[0], by1 = bf[1], bx2 = bf[2], by2 = bf[3];
  const int   blab = bi[5];
  const int   active = bi[6];
  const float barea = (bx2 - bx1) * (by2 - by1);

  float lmax = -3.0e38f;
  int   lorig = 0x7fffffff;
  int   lpos = -1;
  for (unsigned i = blockIdx.x * blockDim.x + threadIdx.x; i < K;
       i += (unsigned)gridDim.x * blockDim.x) {
    float s = csc[i];
    if (active && s > -0.5f && clab[i] == blab) {
      const float4 b = cbx[i];
      const float a2 = (b.z - b.x) * (b.w - b.y);
      const float ix1 = fmaxf(bx1, b.x), iy1 = fmaxf(by1, b.y);
      const float ix2 = fminf(bx2, b.z), iy2 = fminf(by2, b.w);
      const float iw = fmaxf(ix2 - ix1, 0.0f);
      const float ih = fmaxf(iy2 - iy1, 0.0f);
      const float inter = iw * ih;
      const float iou = inter / (barea + a2 - inter);
      if (iou > NMS_THRESH) { s = -1.0f; csc[i] = -1.0f; }
    }
    const int o = cor[i];
    if (s > lmax || (s == lmax && o < lorig)) { lmax = s; lorig = o; lpos = (int)i; }
  }

  __shared__ float sm[NMS_THREADS];
  __shared__ int   so[NMS_THREADS];
  __shared__ int   sp[NMS_THREADS];
  const int tid = threadIdx.x;
  sm[tid] = lmax; so[tid] = lorig; sp[tid] = lpos;
  __syncthreads();
  for (int off = NMS_THREADS / 2; off > 0; off >>= 1) {
    if (tid < off) {
      if (sm[tid + off] > sm[tid] ||
          (sm[tid + off] == sm[tid] && so[tid + off] < so[tid])) {
        sm[tid] = sm[tid + off]; so[tid] = so[tid + off]; sp[tid] = sp[tid + off];
      }
    }
    __syncthreads();
  }
  if (tid == 0) {
    ((float*)(ws + WS_PS))[blockIdx.x]   = sm[0];
    ((int*)(ws + WS_PORIG))[blockIdx.x]  = so[0];
    ((int*)(ws + WS_PPOS))[blockIdx.x]   = sp[0];
  }
}

// ---------------------------------------------------------------------------
// NMS round, part 2: reduce partials, emit output slot, update best state.
// ---------------------------------------------------------------------------
__global__ __launch_bounds__(NMS_THREADS) void nms_pick(unsigned char* ws,
                                                        float* __restrict__ out,
                                                        int slot) {
  const float* ps = (const float*)(ws + WS_PS);
  const int*   po = (const int*)(ws + WS_PORIG);
  const int*   pp = (const int*)(ws + WS_PPOS);

  float lmax = -3.0e38f;
  int   lorig = 0x7fffffff;
  int   lpos = -1;
  for (int i = threadIdx.x; i < NMS_BLOCKS; i += NMS_THREADS) {
    const float s = ps[i];
    const int   o = po[i];
    if (s > lmax || (s == lmax && o < lorig)) { lmax = s; lorig = o; lpos = pp[i]; }
  }
  __shared__ float sm[NMS_THREADS];
  __shared__ int   so[NMS_THREADS];
  __shared__ int   sp[NMS_THREADS];
  const int tid = threadIdx.x;
  sm[tid] = lmax; so[tid] = lorig; sp[tid] = lpos;
  __syncthreads();
  for (int off = NMS_THREADS / 2; off > 0; off >>= 1) {
    if (tid < off) {
      if (sm[tid + off] > sm[tid] ||
          (sm[tid + off] == sm[tid] && so[tid + off] < so[tid])) {
        sm[tid] = sm[tid + off]; so[tid] = so[tid + off]; sp[tid] = sp[tid + off];
      }
    }
    __syncthreads();
  }
  if (tid == 0) {
    float* bf = (float*)(ws + WS_BEST);
    int*   bi = (int*)(ws + WS_BEST);
    float* csc = (float*)(ws + WS_CSCORE);
    const float4* cbx = (const float4*)(ws + WS_CBOX);
    const int*    clb = (const int*)(ws + WS_CLAB);
    const float bscore = sm[0];
    const int   bpos   = sp[0];
    if (bscore > 0.0f && bpos >= 0) {
      const float4 b = cbx[bpos];
      const int lab = clb[bpos];
      out[slot * 4 + 0] = b.x; out[slot * 4 + 1] = b.y;
      out[slot * 4 + 2] = b.z; out[slot * 4 + 3] = b.w;
      out[4 * DETS + slot] = bscore;
      out[5 * DETS + slot] = (float)lab;
      csc[bpos] = -1.0f;                    // pop winner
      bf[0] = b.x; bf[1] = b.y; bf[2] = b.z; bf[3] = b.w;
      bi[5] = lab; bi[6] = 1;
    } else {                                // exhausted -> padded slot
      out[slot * 4 + 0] = 0.0f; out[slot * 4 + 1] = 0.0f;
      out[slot * 4 + 2] = 0.0f; out[slot * 4 + 3] = 0.0f;
      out[4 * DETS + slot] = 0.0f;
      out[5 * DETS + slot] = -1.0f;
      bi[6] = 0;
    }
  }
}

__global__ void nms_init(unsigned char* ws) {
  if (threadIdx.x == 0) {
    *(unsigned*)(ws + WS_CNT) = 0u;
    ((int*)(ws + WS_BEST))[6] = 0;     // active = 0
  }
}

extern "C" void kernel_launch(void* const* d_in, const int* in_sizes, int n_in,
                              void* d_out, int out_size, void* d_ws, size_t ws_size,
                              hipStream_t stream) {
  (void)n_in; (void)out_size; (void)ws_size;
  const float* logits = (const float*)d_in[0];
  const float* breg   = (const float*)d_in[1];
  const float* props  = (const float*)d_in[2];
  const int*   img_h  = (const int*)d_in[3];
  const int*   img_w  = (const int*)d_in[4];
  float* out = (float*)d_out;
  unsigned char* ws = (unsigned char*)d_ws;

  const int N = in_sizes[1] / 4;                       // box_regression is [N,4]
  const int numTiles = (N + TILE_ROWS - 1) / TILE_ROWS;
  const int fgrid = numTiles < 1024 ? numTiles : 1024;

  nms_init<<<1, 64, 0, stream>>>(ws);
  fused_decode_softmax<<<fgrid, FUSED_BLOCK, 0, stream>>>(
      logits, breg, props, img_h, img_w, ws, N, numTiles);
  for (int i = 0; i < DETS; ++i) {
    nms_sweep<<<NMS_BLOCKS, NMS_THREADS, 0, stream>>>(ws);
    nms_pick<<<1, NMS_THREADS, 0, stream>>>(ws, out, i);
  }
}